// _OdachiEngine_83348135346479
// MI455X (gfx1250) — compile-verified
//
#include <hip/hip_runtime.h>
#include <hip/hip_bf16.h>
#include <math.h>

typedef __attribute__((ext_vector_type(2))) float v2f;
typedef __attribute__((ext_vector_type(8))) float v8f;

#define EMBED 41
#define HPAD  48
#define HID   256

// ---- pad atom features [n x 41] -> h [n x 48] (cols 41..47 = 0) ----
__global__ void k_pad_h0(const float* __restrict__ atom, float* __restrict__ h, int n) {
    int idx = blockIdx.x * 256 + threadIdx.x;
    if (idx >= n * HPAD) return;
    int i = idx / HPAD, e = idx % HPAD;
    h[idx] = (e < EMBED) ? atom[i * EMBED + e] : 0.f;
}

// ---- pad W1 [82 x 256] -> W1p [96 x 256]: rows 0..40 = top, rows 48..88 = bottom ----
__global__ void k_pad_w1(const float* __restrict__ W1, float* __restrict__ W1p) {
    int idx = blockIdx.x * 256 + threadIdx.x;   // 96*256 threads
    int r = idx / HID, c = idx % HID;
    float v = 0.f;
    if (r < EMBED)                    v = W1[r * HID + c];
    else if (r >= 48 && r < 48+EMBED) v = W1[(r - 48 + EMBED) * HID + c];
    W1p[idx] = v;
}

// ---- S = adj @ h, f32 WMMA 16x16x4. grid (n/16, 3), block = 1 wave (32) ----
__global__ void k_adj_mm(const float* __restrict__ adj, const float* __restrict__ h,
                         float* __restrict__ S, int n) {
    int lane = threadIdx.x;
    int half = lane >> 4;       // 0: K=0,1  1: K=2,3
    int lr   = lane & 15;
    int r0 = blockIdx.x * 16;
    int c0 = blockIdx.y * 16;
    const float* arow = adj + (size_t)(r0 + lr) * (size_t)n;
    v8f acc = {};
    for (int k = 0; k < n; k += 4) {
        int kk = k + 2 * half;
        v2f a; a.x = arow[kk];                  a.y = arow[kk + 1];
        v2f b; b.x = h[kk * HPAD + c0 + lr];    b.y = h[(kk + 1) * HPAD + c0 + lr];
        acc = __builtin_amdgcn_wmma_f32_16x16x4_f32(false, a, false, b,
                                                    (short)0, acc, false, false);
    }
#pragma unroll
    for (int v = 0; v < 8; ++v)
        S[(size_t)(r0 + v + 8 * half) * HPAD + c0 + lr] = acc[v];
}

// ---- h' = relu(S @ Wg[l]), keep zero padding. 1 thread / element ----
__global__ void k_hw(const float* __restrict__ S, const float* __restrict__ Wg,
                     float* __restrict__ hout, int n) {
    int idx = blockIdx.x * 256 + threadIdx.x;
    if (idx >= n * HPAD) return;
    int i = idx / HPAD, e = idx % HPAD;
    float acc = 0.f;
    if (e < EMBED) {
        const float* srow = S + (size_t)i * HPAD;
        for (int d = 0; d < EMBED; ++d) acc += srow[d] * Wg[d * EMBED + e];
        acc = fmaxf(acc, 0.f);
    }
    hout[idx] = acc;
}

// ---- Amat = h @ W1p[0:48], Bmat = h @ W1p[48:96], f32 WMMA. grid (n/16, 32) ----
__global__ void k_embed_w1(const float* __restrict__ h, const float* __restrict__ W1p,
                           float* __restrict__ Amat, float* __restrict__ Bmat, int n) {
    int lane = threadIdx.x;
    int half = lane >> 4;
    int lr   = lane & 15;
    int r0 = blockIdx.x * 16;
    int y  = blockIdx.y;
    int koff = (y < 16) ? 0 : 48;
    int c0   = ((y < 16) ? y : (y - 16)) * 16;
    float* out = (y < 16) ? Amat : Bmat;
    v8f acc = {};
#pragma unroll
    for (int k = 0; k < HPAD; k += 4) {
        int kk = k + 2 * half;
        v2f a; a.x = h[(r0 + lr) * HPAD + kk];          a.y = h[(r0 + lr) * HPAD + kk + 1];
        v2f b; b.x = W1p[(koff + kk) * HID + c0 + lr];  b.y = W1p[(koff + kk + 1) * HID + c0 + lr];
        acc = __builtin_amdgcn_wmma_f32_16x16x4_f32(false, a, false, b,
                                                    (short)0, acc, false, false);
    }
#pragma unroll
    for (int v = 0; v < 8; ++v)
        out[(size_t)(r0 + v + 8 * half) * HID + c0 + lr] = acc[v];
}

// ---- pair classifier: prob = sigmoid(b2 + sum_k relu(A[i,k]+B[j,k]+b1[k]) * W2[k]) ----
#define LSTR 260   // 260 % 64 == 4 -> conflict-free float4 LDS reads across 16 rows
__global__ void k_pairs(const float* __restrict__ Amat, const float* __restrict__ Bmat,
                        const float* __restrict__ b1,  const float* __restrict__ W2,
                        const float* __restrict__ b2,  float* __restrict__ out,
                        int n, long long P) {
    int bi = blockIdx.y, bj = blockIdx.x;
    if (bj < bi) return;                       // triangular: uniform exit, no barrier yet
    __shared__ __align__(16) float As[16 * LSTR];
    __shared__ __align__(16) float Bs[16 * LSTR];
    __shared__ __align__(16) float b1s[HID];
    __shared__ __align__(16) float w2s[HID];
    int t = threadIdx.x;                       // 256 threads
    int i0 = bi * 16, j0 = bj * 16;
#pragma unroll
    for (int v = 0; v < 4; ++v) {              // 1024 float4 per matrix tile
        int q   = v * 256 + t;
        int row = q >> 6;
        int k4  = (q & 63) << 2;
        float4 av = *(const float4*)&Amat[(size_t)(i0 + row) * HID + k4];
        float4 bv = *(const float4*)&Bmat[(size_t)(j0 + row) * HID + k4];
        *(float4*)&As[row * LSTR + k4] = av;
        *(float4*)&Bs[row * LSTR + k4] = bv;
    }
    b1s[t] = b1[t];
    w2s[t] = W2[t];
    __syncthreads();

    int ti = t >> 4, tj = t & 15;
    int i = i0 + ti, j = j0 + tj;
    const float* ar = &As[ti * LSTR];
    const float* br = &Bs[tj * LSTR];
    float acc = 0.f;
#pragma unroll 4
    for (int k = 0; k < HID; k += 4) {
        float4 av  = *(const float4*)&ar[k];
        float4 bv  = *(const float4*)&br[k];
        float4 b1v = *(const float4*)&b1s[k];
        float4 wv  = *(const float4*)&w2s[k];
        acc += fmaxf(av.x + bv.x + b1v.x, 0.f) * wv.x;
        acc += fmaxf(av.y + bv.y + b1v.y, 0.f) * wv.y;
        acc += fmaxf(av.z + bv.z + b1v.z, 0.f) * wv.z;
        acc += fmaxf(av.w + bv.w + b1v.w, 0.f) * wv.w;
    }
    if (i < j) {
        float x  = acc + b2[0];
        float pr = 1.f / (1.f + __expf(-x));
        long long p = (long long)i * (2LL * n - i - 1) / 2 + (long long)(j - i - 1);
        float2 ij = make_float2((float)i, (float)j);
        *(float2*)&out[2 * p] = ij;            // idxs, row-major [P,2]
        out[2 * P + p] = pr;                   // probs [P,1] after idxs
    }
}

extern "C" void kernel_launch(void* const* d_in, const int* in_sizes, int n_in,
                              void* d_out, int out_size, void* d_ws, size_t ws_size,
                              hipStream_t stream) {
    const float* adj  = (const float*)d_in[0];
    const float* atom = (const float*)d_in[1];
    const float* Wg   = (const float*)d_in[2];
    const float* W1   = (const float*)d_in[3];
    const float* b1   = (const float*)d_in[4];
    const float* W2   = (const float*)d_in[5];
    const float* b2   = (const float*)d_in[6];
    float* out = (float*)d_out;

    int n = in_sizes[1] / EMBED;               // 2048
    long long P = (long long)n * (n - 1) / 2;

    float* ws   = (float*)d_ws;
    float* hA   = ws;
    float* hB   = hA   + (size_t)n * HPAD;
    float* S    = hB   + (size_t)n * HPAD;
    float* W1p  = S    + (size_t)n * HPAD;
    float* Amat = W1p  + (size_t)96 * HID;
    float* Bmat = Amat + (size_t)n * HID;      // total ~5.5 MB of d_ws

    int nh = n * HPAD;
    k_pad_h0<<<(nh + 255) / 256, 256, 0, stream>>>(atom, hA, n);
    k_pad_w1<<<(96 * HID) / 256, 256, 0, stream>>>(W1, W1p);

    float* cur = hA; float* nxt = hB;
    for (int l = 0; l < 4; ++l) {
        k_adj_mm<<<dim3(n / 16, 3), 32, 0, stream>>>(adj, cur, S, n);
        k_hw<<<(nh + 255) / 256, 256, 0, stream>>>(S, Wg + (size_t)l * EMBED * EMBED, nxt, n);
        float* tmp = cur; cur = nxt; nxt = tmp;
    }

    k_embed_w1<<<dim3(n / 16, 32), 32, 0, stream>>>(cur, W1p, Amat, Bmat, n);
    k_pairs<<<dim3(n / 16, n / 16), 256, 0, stream>>>(Amat, Bmat, b1, W2, b2, out, n, P);
}